// StackedMamba_75823352643797
// MI455X (gfx1250) — compile-verified
//
#include <hip/hip_runtime.h>

// Problem constants (from reference)
#define DM   1024          // d_model
#define DI   2048          // d_inner
#define DS   16            // d_state
#define DTR  64            // dt_rank
#define NXD  96            // dt_rank + 2*d_state
#define BB   4
#define LL   2048
#define ROWS (BB*LL)       // 8192
#define NL   4

typedef __attribute__((ext_vector_type(2))) float v2f;
typedef __attribute__((ext_vector_type(8))) float v8f;

__device__ __forceinline__ float sigmoidf_(float x) { return 1.0f / (1.0f + __expf(-x)); }

// ---------------- LayerNorm: one 256-thread block per row of 1024 (float4 I/O) ----------------
__global__ void ln_kernel(const float* __restrict__ x, const float* __restrict__ w,
                          const float* __restrict__ b, float* __restrict__ out)
{
    __shared__ float sbuf[256];
    const int row = blockIdx.x;
    const int tid = threadIdx.x;
    float4 v = ((const float4*)(x + (size_t)row * DM))[tid];
    float s = v.x + v.y + v.z + v.w;
    sbuf[tid] = s; __syncthreads();
    for (int off = 128; off > 0; off >>= 1) { if (tid < off) sbuf[tid] += sbuf[tid + off]; __syncthreads(); }
    const float mu = sbuf[0] * (1.0f / DM);
    __syncthreads();
    v.x -= mu; v.y -= mu; v.z -= mu; v.w -= mu;
    sbuf[tid] = v.x * v.x + v.y * v.y + v.z * v.z + v.w * v.w; __syncthreads();
    for (int off = 128; off > 0; off >>= 1) { if (tid < off) sbuf[tid] += sbuf[tid + off]; __syncthreads(); }
    const float rstd = rsqrtf(sbuf[0] * (1.0f / DM) + 1e-5f);
    const float4 wv = ((const float4*)w)[tid];
    const float4 bv = ((const float4*)b)[tid];
    float4 o;
    o.x = v.x * rstd * wv.x + bv.x;
    o.y = v.y * rstd * wv.y + bv.y;
    o.z = v.z * rstd * wv.z + bv.z;
    o.w = v.w * rstd * wv.w + bv.w;
    ((float4*)(out + (size_t)row * DM))[tid] = o;
}

// ---------------- fp32 WMMA GEMM:  D[MxN](ldd) = A[MxK](lda) @ W[NxK](ldw)^T ----------------
// Block = 128 threads = 4 waves stacked in M; block tile 128x64, wave tile 32x64.
// Register double-buffered K pipeline: prefetch k+4 fragments before the k WMMAs so the
// scheduler can overlap global_load latency with v_wmma issue (no loadcnt-0 stalls).
// 32-bit element offsets off the uniform base -> saddr-form global loads, minimal VALU.
// EPI: 0 = none; 1 = add residual R (same shape/ldd); 2 = softplus(v + R[n]) (per-col bias)
template <int EPI>
__global__ void __launch_bounds__(128)
gemm_wmma(const float* __restrict__ A, const float* __restrict__ W,
          float* __restrict__ D, const float* __restrict__ R,
          int N, int K, int lda, int ldw, int ldd)
{
    const int lane = threadIdx.x & 31;
    const int wave = threadIdx.x >> 5;
    const int lm   = lane & 15;   // row (A) / col (B,D) within 16
    const int hi   = lane >> 4;   // K pair {0,1} vs {2,3}; row-half for D
    const int row0 = blockIdx.y * 128 + wave * 32;
    const int col0 = blockIdx.x * 64;

    const v8f zero = {0.f, 0.f, 0.f, 0.f, 0.f, 0.f, 0.f, 0.f};
    v8f acc[2][4];
#pragma unroll
    for (int m = 0; m < 2; ++m)
#pragma unroll
        for (int j = 0; j < 4; ++j) acc[m][j] = zero;

    // 32-bit element offsets (all buffers < 2^31 elements)
    const unsigned aoff0 = (unsigned)(row0 + lm) * (unsigned)lda + 2u * hi;
    const unsigned aoff1 = aoff0 + 16u * (unsigned)lda;
    unsigned boff[4];
#pragma unroll
    for (int j = 0; j < 4; ++j) {
        int n = col0 + j * 16 + lm;
        n = (n < N) ? n : (N - 1);            // branchless clamp; dead columns never stored
        boff[j] = (unsigned)n * (unsigned)ldw + 2u * hi;
    }

    // prologue: fragments for k = 0
    v2f ca0 = *(const v2f*)(A + aoff0);
    v2f ca1 = *(const v2f*)(A + aoff1);
    v2f cb[4];
#pragma unroll
    for (int j = 0; j < 4; ++j) cb[j] = *(const v2f*)(W + boff[j]);

    for (int k = 4; k < K; k += 4) {
        // prefetch k (next) fragments
        v2f na0 = *(const v2f*)(A + aoff0 + k);
        v2f na1 = *(const v2f*)(A + aoff1 + k);
        v2f nb[4];
#pragma unroll
        for (int j = 0; j < 4; ++j) nb[j] = *(const v2f*)(W + boff[j] + k);
        // compute k-4 (current) fragments
#pragma unroll
        for (int j = 0; j < 4; ++j) {
            acc[0][j] = __builtin_amdgcn_wmma_f32_16x16x4_f32(
                false, ca0, false, cb[j], (short)0, acc[0][j], false, false);
            acc[1][j] = __builtin_amdgcn_wmma_f32_16x16x4_f32(
                false, ca1, false, cb[j], (short)0, acc[1][j], false, false);
        }
        ca0 = na0; ca1 = na1;
#pragma unroll
        for (int j = 0; j < 4; ++j) cb[j] = nb[j];
    }
    // epilogue K-step
#pragma unroll
    for (int j = 0; j < 4; ++j) {
        acc[0][j] = __builtin_amdgcn_wmma_f32_16x16x4_f32(
            false, ca0, false, cb[j], (short)0, acc[0][j], false, false);
        acc[1][j] = __builtin_amdgcn_wmma_f32_16x16x4_f32(
            false, ca1, false, cb[j], (short)0, acc[1][j], false, false);
    }

#pragma unroll
    for (int m = 0; m < 2; ++m) {
#pragma unroll
        for (int j = 0; j < 4; ++j) {
            const int n = col0 + j * 16 + lm;
            if (n >= N) continue;
#pragma unroll
            for (int r = 0; r < 8; ++r) {
                const int row = row0 + m * 16 + hi * 8 + r;
                const size_t off = (size_t)row * ldd + n;
                float vv = acc[m][j][r];
                if (EPI == 1) vv += R[off];
                if (EPI == 2) { vv += R[n]; vv = (vv > 20.f) ? vv : log1pf(expf(vv)); }
                D[off] = vv;
            }
        }
    }
}

// ---------------- causal depthwise conv (width 4) + SiLU, float4 over channels ----------------
__global__ void conv_silu_kernel(const float* __restrict__ xz, const float* __restrict__ cw,
                                 const float* __restrict__ cb, float* __restrict__ u)
{
    const size_t idx4 = (size_t)blockIdx.x * 256 + threadIdx.x;   // over ROWS*DI/4
    const int d4 = (int)(idx4 & (DI / 4 - 1)) * 4;                // channel group
    const long long bt = (long long)(idx4 >> 9);                  // row index (b*L + t)
    const int t = (int)(bt & (LL - 1));
    const long long str = 2 * DI;
    const float* p = xz + bt * str + d4;

    // taps: xv[j] = xs[t-3+j][d4..d4+3]
    const float4 zero4 = {0.f, 0.f, 0.f, 0.f};
    float4 xv0 = (t >= 3) ? *(const float4*)(p - 3 * str) : zero4;
    float4 xv1 = (t >= 2) ? *(const float4*)(p - 2 * str) : zero4;
    float4 xv2 = (t >= 1) ? *(const float4*)(p - 1 * str) : zero4;
    float4 xv3 = *(const float4*)(p);
    // per-channel taps: cw[(d4+c)*4 + j]
    float4 c0 = *(const float4*)(cw + (size_t)(d4 + 0) * 4);
    float4 c1 = *(const float4*)(cw + (size_t)(d4 + 1) * 4);
    float4 c2 = *(const float4*)(cw + (size_t)(d4 + 2) * 4);
    float4 c3 = *(const float4*)(cw + (size_t)(d4 + 3) * 4);
    float4 acc = *(const float4*)(cb + d4);
    acc.x += c0.x * xv0.x + c0.y * xv1.x + c0.z * xv2.x + c0.w * xv3.x;
    acc.y += c1.x * xv0.y + c1.y * xv1.y + c1.z * xv2.y + c1.w * xv3.y;
    acc.z += c2.x * xv0.z + c2.y * xv1.z + c2.z * xv2.z + c2.w * xv3.z;
    acc.w += c3.x * xv0.w + c3.y * xv1.w + c3.z * xv2.w + c3.w * xv3.w;
    float4 o;
    o.x = acc.x * sigmoidf_(acc.x);
    o.y = acc.y * sigmoidf_(acc.y);
    o.z = acc.z * sigmoidf_(acc.z);
    o.w = acc.w * sigmoidf_(acc.w);
    *(float4*)(u + (size_t)idx4 * 4) = o;
}

// ---------------- selective scan: one thread per (b, d) channel ----------------
__global__ void scan_kernel(const float* __restrict__ dt, const float* __restrict__ xdbl,
                            const float* __restrict__ u, const float* __restrict__ Alog,
                            float* __restrict__ ys)
{
    const int gid = blockIdx.x * 256 + threadIdx.x;  // 0 .. BB*DI-1
    const int d = gid & (DI - 1);
    const int b = gid >> 11;
    float a[DS], h[DS];
#pragma unroll
    for (int s = 0; s < DS; ++s) { a[s] = -__expf(Alog[d * DS + s]); h[s] = 0.f; }
    const float* dtp = dt + (size_t)b * LL * DI + d;
    const float* up  = u  + (size_t)b * LL * DI + d;
    const float* xd  = xdbl + (size_t)b * LL * NXD + DTR;  // -> [B(16) | C(16)]
    float* yp = ys + (size_t)b * LL * DI + d;
    for (int t = 0; t < LL; ++t) {
        const float dtv = dtp[(size_t)t * DI];
        const float du  = dtv * up[(size_t)t * DI];
        const float* bc = xd + (size_t)t * NXD;  // wave-uniform broadcast loads
        float y = 0.f;
#pragma unroll
        for (int s = 0; s < DS; ++s) {
            const float dA = __expf(dtv * a[s]);
            h[s] = dA * h[s] + du * bc[s];
            y = fmaf(h[s], bc[DS + s], y);
        }
        yp[(size_t)t * DI] = y;
    }
}

// ---------------- gate: ys = (ys + u*Dp) * silu(z), float4 ----------------
__global__ void combine_kernel(float* __restrict__ ys, const float* __restrict__ u,
                               const float* __restrict__ xz, const float* __restrict__ Dp)
{
    const size_t idx4 = (size_t)blockIdx.x * 256 + threadIdx.x;  // over ROWS*DI/4
    const int d4 = (int)(idx4 & (DI / 4 - 1)) * 4;
    const size_t row = idx4 >> 9;
    const float4 z  = *(const float4*)(xz + row * (size_t)(2 * DI) + DI + d4);
    const float4 dp = *(const float4*)(Dp + d4);
    const float4 uv = *(const float4*)(u + idx4 * 4);
    float4 yv = *(const float4*)(ys + idx4 * 4);
    yv.x = (yv.x + uv.x * dp.x) * (z.x * sigmoidf_(z.x));
    yv.y = (yv.y + uv.y * dp.y) * (z.y * sigmoidf_(z.y));
    yv.z = (yv.z + uv.z * dp.z) * (z.z * sigmoidf_(z.z));
    yv.w = (yv.w + uv.w * dp.w) * (z.w * sigmoidf_(z.w));
    *(float4*)(ys + idx4 * 4) = yv;
}

extern "C" void kernel_launch(void* const* d_in, const int* in_sizes, int n_in,
                              void* d_out, int out_size, void* d_ws, size_t ws_size,
                              hipStream_t stream)
{
    const float* x_in   = (const float*)d_in[0];
    const float* W_in   = (const float*)d_in[1];
    const float* conv_w = (const float*)d_in[2];
    const float* conv_b = (const float*)d_in[3];
    const float* W_x    = (const float*)d_in[4];
    const float* W_dt   = (const float*)d_in[5];
    const float* b_dt   = (const float*)d_in[6];
    const float* A_log  = (const float*)d_in[7];
    const float* Dp     = (const float*)d_in[8];
    const float* W_out  = (const float*)d_in[9];
    const float* ln_w   = (const float*)d_in[10];
    const float* ln_b   = (const float*)d_in[11];
    float* out = (float*)d_out;

    // Workspace carve-up (~372 MB)
    float* xn   = (float*)d_ws;                 // ROWS*DM
    float* xz   = xn   + (size_t)ROWS * DM;     // ROWS*2*DI
    float* u    = xz   + (size_t)ROWS * 2 * DI; // ROWS*DI
    float* xdbl = u    + (size_t)ROWS * DI;     // ROWS*NXD
    float* dt   = xdbl + (size_t)ROWS * NXD;    // ROWS*DI
    float* ys   = dt   + (size_t)ROWS * DI;     // ROWS*DI

    for (int i = 0; i < NL; ++i) {
        const float* xin = (i == 0) ? x_in : out;

        ln_kernel<<<ROWS, 256, 0, stream>>>(xin, ln_w + (size_t)i * DM, ln_b + (size_t)i * DM, xn);

        // xz = xn @ W_in^T   (M=8192, N=4096, K=1024)
        gemm_wmma<0><<<dim3(2 * DI / 64, ROWS / 128), 128, 0, stream>>>(
            xn, W_in + (size_t)i * 2 * DI * DM, xz, nullptr, 2 * DI, DM, DM, DM, 2 * DI);

        conv_silu_kernel<<<(size_t)ROWS * DI / 4 / 256, 256, 0, stream>>>(
            xz, conv_w + (size_t)i * DI * 4, conv_b + (size_t)i * DI, u);

        // xdbl = u @ W_x^T   (M=8192, N=96, K=2048)
        gemm_wmma<0><<<dim3(2, ROWS / 128), 128, 0, stream>>>(
            u, W_x + (size_t)i * NXD * DI, xdbl, nullptr, NXD, DI, DI, DI, NXD);

        // dt = softplus(xdbl[:, :64] @ W_dt^T + b_dt)   (M=8192, N=2048, K=64)
        gemm_wmma<2><<<dim3(DI / 64, ROWS / 128), 128, 0, stream>>>(
            xdbl, W_dt + (size_t)i * DI * DTR, dt, b_dt + (size_t)i * DI, DI, DTR, NXD, DTR, DI);

        scan_kernel<<<BB * DI / 256, 256, 0, stream>>>(
            dt, xdbl, u, A_log + (size_t)i * DI * DS, ys);

        combine_kernel<<<(size_t)ROWS * DI / 4 / 256, 256, 0, stream>>>(
            ys, u, xz, Dp + (size_t)i * DI);

        // out = ys @ W_out^T + xin   (M=8192, N=1024, K=2048); in-place residual is per-element RMW
        gemm_wmma<1><<<dim3(DM / 64, ROWS / 128), 128, 0, stream>>>(
            ys, W_out + (size_t)i * DM * DI, out, xin, DM, DI, DI, DI, DM);
    }
}